// Attention_6528350290412
// MI455X (gfx1250) — compile-verified
//
#include <hip/hip_runtime.h>
#include <math.h>

// Problem constants (PatchTST attention block)
// B=16, N=256, P=8, D=512, H=8, E=P*D=4096, dh=E/H=512
#define BB   16
#define NN   256
#define HH   8
#define EE   4096
#define DH   512
#define MROWS (BB * NN)   // 4096

typedef __attribute__((ext_vector_type(2))) float v2f;
typedef __attribute__((ext_vector_type(8))) float v8f;

// ---------------------------------------------------------------------------
// CDNA5 async global->LDS copy (ASYNCcnt-tracked). The builtin's params are
// v4i pointers in explicit address spaces (confirmed by compiler diagnostic):
//   (v4i addrspace(1)* src_global, v4i addrspace(3)* dst_lds, imm off, imm cpol)
// Falls back to a plain load/store pair if the builtin isn't declared.
// ---------------------------------------------------------------------------
#if defined(__has_builtin)
#if __has_builtin(__builtin_amdgcn_global_load_async_to_lds_b128)
#define HAVE_ASYNC_B128 1
#endif
#if __has_builtin(__builtin_amdgcn_s_wait_asynccnt)
#define HAVE_WAIT_ASYNC 1
#endif
#endif

typedef int v4i __attribute__((vector_size(4 * sizeof(int))));
typedef __attribute__((address_space(1))) v4i gas_v4i;   // global
typedef __attribute__((address_space(3))) v4i las_v4i;   // LDS

__device__ __forceinline__ void cp16(float* dst_lds, const float* src_g)
{
#ifdef HAVE_ASYNC_B128
    __builtin_amdgcn_global_load_async_to_lds_b128(
        (gas_v4i*)src_g, (las_v4i*)dst_lds, 0, 0);
#else
    *(float4*)dst_lds = *(const float4*)src_g;
#endif
}

__device__ __forceinline__ void wait_async_all()
{
#ifdef HAVE_ASYNC_B128
#ifdef HAVE_WAIT_ASYNC
    __builtin_amdgcn_s_wait_asynccnt(0);
#else
    asm volatile("s_wait_asynccnt 0x0" ::: "memory");
#endif
#endif
}

// ---------------------------------------------------------------------------
// GEMM: Y[M x Nd] = X[M x Kd] @ W^T + bias (+ optional pos-embed add)
// X row-major [M x Kd], W row-major [Nd x Kd] (so B[k][n] = W[n][k], which is
// k-contiguous per n -> same LDS staging pattern as X).
// Tile: 128x128, K-block 32, double-buffered LDS (one barrier per K-step):
// async copies for tile k+1 are issued right after the barrier and overlap
// the WMMA consumption of tile k. 8 wave32 waves; each wave owns a 64x32
// patch = 4x2 grid of 16x16 V_WMMA_F32_16X16X4_F32 accumulators.
// ---------------------------------------------------------------------------
__global__ __launch_bounds__(256)
void gemm_xwT(const float* __restrict__ X, const float* __restrict__ W,
              const float* __restrict__ bias, const float* __restrict__ pos,
              float* __restrict__ Y, int Kd, int Nd)
{
    __shared__ float As[2][128 * 32];   // X tile  [row][k]
    __shared__ float Bs[2][128 * 32];   // W tile  [n][k]

    const int tid   = threadIdx.x;
    const int lane  = tid & 31;          // wave32
    const int w     = tid >> 5;          // 8 waves
    const int half  = lane >> 4;         // 0/1 (K-pair select / M-half select)
    const int l16   = lane & 15;

    const int m_blk = blockIdx.x * 128;
    const int n_blk = blockIdx.y * 128;
    const int wm    = (w >> 2) * 64;     // wave row base within tile
    const int wn    = (w & 3) * 32;      // wave col base within tile

    v8f acc[4][2];
    #pragma unroll
    for (int i = 0; i < 4; ++i)
        #pragma unroll
        for (int j = 0; j < 2; ++j)
            #pragma unroll
            for (int r = 0; r < 8; ++r) acc[i][j][r] = 0.0f;

    // staging: 128 rows x 32 floats per tile; 8 threads/row (b128 each),
    // 4 row-passes of 32 rows.
    const int arow  = tid >> 3;          // 0..31
    const int acol4 = (tid & 7) * 4;     // 0,4,...,28
    const float* Xrow = X + (size_t)(m_blk + arow) * Kd + acol4;
    const float* Wrow = W + (size_t)(n_blk + arow) * Kd + acol4;

    // prologue: tile 0 -> buffer 0
    #pragma unroll
    for (int i = 0; i < 4; ++i) {
        cp16(&As[0][(arow + i * 32) * 32 + acol4], Xrow + (size_t)i * 32 * Kd);
        cp16(&Bs[0][(arow + i * 32) * 32 + acol4], Wrow + (size_t)i * 32 * Kd);
    }

    int buf = 0;
    for (int k0 = 0; k0 < Kd; k0 += 32) {
        wait_async_all();
        __syncthreads();                 // tile `buf` resident in LDS

        if (k0 + 32 < Kd) {              // stream tile k+1 into the other buffer
            #pragma unroll
            for (int i = 0; i < 4; ++i) {
                cp16(&As[buf ^ 1][(arow + i * 32) * 32 + acol4],
                     Xrow + (size_t)i * 32 * Kd + (k0 + 32));
                cp16(&Bs[buf ^ 1][(arow + i * 32) * 32 + acol4],
                     Wrow + (size_t)i * 32 * Kd + (k0 + 32));
            }
        }

        const float* Ab = As[buf];
        const float* Bb = Bs[buf];
        #pragma unroll
        for (int kk = 0; kk < 32; kk += 4) {
            const int kidx = kk + half * 2;          // this lane's K pair
            v2f a[4], b2[2];
            #pragma unroll
            for (int mt = 0; mt < 4; ++mt)
                a[mt] = *(const v2f*)(&Ab[(wm + mt * 16 + l16) * 32 + kidx]);
            #pragma unroll
            for (int nt = 0; nt < 2; ++nt)
                b2[nt] = *(const v2f*)(&Bb[(wn + nt * 16 + l16) * 32 + kidx]);
            #pragma unroll
            for (int mt = 0; mt < 4; ++mt)
                #pragma unroll
                for (int nt = 0; nt < 2; ++nt)
                    acc[mt][nt] = __builtin_amdgcn_wmma_f32_16x16x4_f32(
                        false, a[mt], false, b2[nt], (short)0, acc[mt][nt], false, false);
        }
        buf ^= 1;
    }

    // Epilogue: D layout -> m = base + r + half*8, n = base + (lane&15)
    #pragma unroll
    for (int mt = 0; mt < 4; ++mt) {
        #pragma unroll
        for (int nt = 0; nt < 2; ++nt) {
            #pragma unroll
            for (int r = 0; r < 8; ++r) {
                const int m = m_blk + wm + mt * 16 + r + half * 8;
                const int n = n_blk + wn + nt * 16 + l16;
                float val = acc[mt][nt][r] + bias[n];
                if (pos) val += pos[(size_t)(m & (NN - 1)) * EE + n];
                Y[(size_t)m * Nd + n] = val;
            }
        }
    }
}

// ---------------------------------------------------------------------------
// Fused attention scores + softmax.
// grid: x = m-tile (N/64 = 4), y = b*H + h. Block 256 threads / 8 waves.
// scores[64 x 256] = Q_bh[64 x 512] @ K_bh^T * (1/8), then row softmax,
// written directly into the attn_weights output region of d_out.
// LDS: 64KB buffer; first 40KB used as Q/K tiles during the K loop, whole
// buffer reused as the 64x256 score tile for softmax. (~1% of total FLOPs,
// kept single-buffered.)
// ---------------------------------------------------------------------------
__global__ __launch_bounds__(256)
void attn_scores_softmax(const float* __restrict__ Q, const float* __restrict__ Km,
                         float* __restrict__ attn)
{
    __shared__ float smem[16384];        // 64 KB
    float* As = smem;                    // Q tile: 64 x 32   (2048 floats)
    float* Bs = smem + 2048;             // K tile: 256 x 32  (8192 floats)

    const int tid  = threadIdx.x;
    const int lane = tid & 31;
    const int w    = tid >> 5;
    const int half = lane >> 4;
    const int l16  = lane & 15;

    const int bh = blockIdx.y;
    const int b  = bh >> 3;
    const int h  = bh & 7;
    const int m0 = blockIdx.x * 64;

    const int wm = (w >> 2) * 32;        // 2 m-tiles per wave (32 rows)
    const int wn = (w & 3) * 64;         // 4 n-tiles per wave (64 cols)

    v8f acc[2][4];
    #pragma unroll
    for (int i = 0; i < 2; ++i)
        #pragma unroll
        for (int j = 0; j < 4; ++j)
            #pragma unroll
            for (int r = 0; r < 8; ++r) acc[i][j][r] = 0.0f;

    const int arow  = tid >> 3;
    const int acol4 = (tid & 7) * 4;

    const size_t qbase = ((size_t)b * NN) * EE + (size_t)h * DH;

    for (int k0 = 0; k0 < DH; k0 += 32) {
        // Q tile: 64 rows x 32
        #pragma unroll
        for (int i = 0; i < 2; ++i) {
            const int r = arow + i * 32;
            const float4 vq = *(const float4*)(Q + qbase + (size_t)(m0 + r) * EE + k0 + acol4);
            *(float4*)(&As[r * 32 + acol4]) = vq;
        }
        // K tile: 256 rows x 32
        #pragma unroll
        for (int i = 0; i < 8; ++i) {
            const int r = arow + i * 32;
            const float4 vk = *(const float4*)(Km + qbase + (size_t)r * EE + k0 + acol4);
            *(float4*)(&Bs[r * 32 + acol4]) = vk;
        }
        __syncthreads();

        #pragma unroll
        for (int kk = 0; kk < 32; kk += 4) {
            const int kidx = kk + half * 2;
            v2f a[2], b2[4];
            #pragma unroll
            for (int mt = 0; mt < 2; ++mt)
                a[mt] = *(const v2f*)(&As[(wm + mt * 16 + l16) * 32 + kidx]);
            #pragma unroll
            for (int nt = 0; nt < 4; ++nt)
                b2[nt] = *(const v2f*)(&Bs[(wn + nt * 16 + l16) * 32 + kidx]);
            #pragma unroll
            for (int mt = 0; mt < 2; ++mt)
                #pragma unroll
                for (int nt = 0; nt < 4; ++nt)
                    acc[mt][nt] = __builtin_amdgcn_wmma_f32_16x16x4_f32(
                        false, a[mt], false, b2[nt], (short)0, acc[mt][nt], false, false);
        }
        __syncthreads();
    }

    // Dump scaled scores into LDS [64][256]; scale = 1/sqrt(D/H) = 1/8
    #pragma unroll
    for (int mt = 0; mt < 2; ++mt)
        #pragma unroll
        for (int nt = 0; nt < 4; ++nt)
            #pragma unroll
            for (int r = 0; r < 8; ++r) {
                const int row = wm + mt * 16 + r + half * 8;
                const int col = wn + nt * 16 + l16;
                smem[row * 256 + col] = acc[mt][nt][r] * 0.125f;
            }
    __syncthreads();

    // Row softmax (64 rows, one thread per row; tiny fraction of total work)
    if (tid < 64) {
        float* rowp = &smem[tid * 256];
        float mx = rowp[0];
        for (int j = 1; j < 256; ++j) mx = fmaxf(mx, rowp[j]);
        float s = 0.0f;
        for (int j = 0; j < 256; ++j) { const float e = __expf(rowp[j] - mx); rowp[j] = e; s += e; }
        const float inv = 1.0f / s;
        float* op = attn + (((size_t)bh * NN) + (m0 + tid)) * NN;
        for (int j = 0; j < 256; ++j) op[j] = rowp[j] * inv;
    }
}

// ---------------------------------------------------------------------------
// Context GEMM: ctx_bh[256 x 512] = attn_bh[256 x 256] @ V_bh[256 x 512]
// grid: x encodes (m-tile 0..1, n-tile 0..3), y = b*H + h.
// B source here is [k][n] with n contiguous -> LDS B tile stored [k][n].
// Double-buffered LDS with async copies, one barrier per K-step.
// ---------------------------------------------------------------------------
__global__ __launch_bounds__(256)
void attn_ctx(const float* __restrict__ attn, const float* __restrict__ V,
              float* __restrict__ ctx)
{
    __shared__ float As[2][128 * 32];   // attn tile [row][k]
    __shared__ float Bs[2][32 * 128];   // V tile    [k][n]

    const int tid  = threadIdx.x;
    const int lane = tid & 31;
    const int w    = tid >> 5;
    const int half = lane >> 4;
    const int l16  = lane & 15;

    const int bh = blockIdx.y;
    const int b  = bh >> 3;
    const int h  = bh & 7;
    const int m0 = (blockIdx.x >> 2) * 128;
    const int n0 = (blockIdx.x & 3) * 128;

    const int wm = (w >> 2) * 64;
    const int wn = (w & 3) * 32;

    v8f acc[4][2];
    #pragma unroll
    for (int i = 0; i < 4; ++i)
        #pragma unroll
        for (int j = 0; j < 2; ++j)
            #pragma unroll
            for (int r = 0; r < 8; ++r) acc[i][j][r] = 0.0f;

    const int arow  = tid >> 3;          // 0..31
    const int acol4 = (tid & 7) * 4;     // 0..28
    const int bkr   = tid >> 5;          // 0..7
    const int bc4   = (tid & 31) * 4;    // 0..124

    const size_t vbase = ((size_t)b * NN) * EE + (size_t)h * DH;
    const size_t abase = (size_t)bh * NN * NN;
    const float* Arow = attn + abase + (size_t)(m0 + arow) * NN + acol4;
    const float* Vrow = V + vbase + (size_t)bkr * EE + n0 + bc4;

    // prologue: tile 0 -> buffer 0
    #pragma unroll
    for (int i = 0; i < 4; ++i) {
        cp16(&As[0][(arow + i * 32) * 32 + acol4], Arow + (size_t)i * 32 * NN);
        cp16(&Bs[0][(bkr + i * 8) * 128 + bc4],    Vrow + (size_t)i * 8 * EE);
    }

    int buf = 0;
    for (int k0 = 0; k0 < NN; k0 += 32) {
        wait_async_all();
        __syncthreads();

        if (k0 + 32 < NN) {
            #pragma unroll
            for (int i = 0; i < 4; ++i) {
                cp16(&As[buf ^ 1][(arow + i * 32) * 32 + acol4],
                     Arow + (size_t)i * 32 * NN + (k0 + 32));
                cp16(&Bs[buf ^ 1][(bkr + i * 8) * 128 + bc4],
                     Vrow + (size_t)(k0 + 32 + i * 8) * EE);
            }
        }

        const float* Ab = As[buf];
        const float* Bb = Bs[buf];
        #pragma unroll
        for (int kk = 0; kk < 32; kk += 4) {
            const int kidx = kk + half * 2;
            v2f a[4], b2[2];
            #pragma unroll
            for (int mt = 0; mt < 4; ++mt)
                a[mt] = *(const v2f*)(&Ab[(wm + mt * 16 + l16) * 32 + kidx]);
            #pragma unroll
            for (int nt = 0; nt < 2; ++nt) {
                const int n = wn + nt * 16 + l16;
                v2f bb;
                bb.x = Bb[kidx * 128 + n];
                bb.y = Bb[(kidx + 1) * 128 + n];
                b2[nt] = bb;
            }
            #pragma unroll
            for (int mt = 0; mt < 4; ++mt)
                #pragma unroll
                for (int nt = 0; nt < 2; ++nt)
                    acc[mt][nt] = __builtin_amdgcn_wmma_f32_16x16x4_f32(
                        false, a[mt], false, b2[nt], (short)0, acc[mt][nt], false, false);
        }
        buf ^= 1;
    }

    #pragma unroll
    for (int mt = 0; mt < 4; ++mt)
        #pragma unroll
        for (int nt = 0; nt < 2; ++nt)
            #pragma unroll
            for (int r = 0; r < 8; ++r) {
                const int m = m0 + wm + mt * 16 + r + half * 8;
                const int n = n0 + wn + nt * 16 + l16;
                ctx[vbase + (size_t)m * EE + n] = acc[mt][nt][r];
            }
}

// ---------------------------------------------------------------------------
extern "C" void kernel_launch(void* const* d_in, const int* in_sizes, int n_in,
                              void* d_out, int out_size, void* d_ws, size_t ws_size,
                              hipStream_t stream)
{
    const float* x       = (const float*)d_in[0];   // (B,N,P,D) == (4096,4096) flat
    const float* pos     = (const float*)d_in[1];   // (256,4096)
    const float* W_embed = (const float*)d_in[2];
    const float* b_embed = (const float*)d_in[3];
    const float* Wq      = (const float*)d_in[4];
    const float* bq      = (const float*)d_in[5];
    const float* Wk      = (const float*)d_in[6];
    const float* bk      = (const float*)d_in[7];
    const float* Wv      = (const float*)d_in[8];
    const float* bv      = (const float*)d_in[9];
    const float* Wo      = (const float*)d_in[10];
    const float* bo      = (const float*)d_in[11];

    float* out  = (float*)d_out;                         // (B,N,P,D) = 16,777,216 floats
    float* attn = out + (size_t)MROWS * EE;              // (B,H,N,N) =  8,388,608 floats

    float* ws  = (float*)d_ws;
    float* xe  = ws;                                     // 16M floats
    float* q   = ws + (size_t)MROWS * EE;                // 16M
    float* k   = ws + 2 * (size_t)MROWS * EE;            // 16M
    float* v   = ws + 3 * (size_t)MROWS * EE;            // 16M
    float* ctx = xe;                                     // reuse xe (dead after QKV)

    const dim3 blk(256);
    const dim3 gG(MROWS / 128, EE / 128);                // 32 x 32

    // xe = x @ W_embed^T + b_embed + pos_embed
    gemm_xwT<<<gG, blk, 0, stream>>>(x,  W_embed, b_embed, pos,     xe,  EE, EE);
    // q/k/v = xe @ W^T + b
    gemm_xwT<<<gG, blk, 0, stream>>>(xe, Wq,      bq,      nullptr, q,   EE, EE);
    gemm_xwT<<<gG, blk, 0, stream>>>(xe, Wk,      bk,      nullptr, k,   EE, EE);
    gemm_xwT<<<gG, blk, 0, stream>>>(xe, Wv,      bv,      nullptr, v,   EE, EE);
    // attn_weights = softmax(q k^T / 8)  -> straight into d_out region 2
    attn_scores_softmax<<<dim3(NN / 64, BB * HH), blk, 0, stream>>>(q, k, attn);
    // ctx = attn @ v
    attn_ctx<<<dim3(8, BB * HH), blk, 0, stream>>>(attn, v, ctx);
    // out = ctx @ Wo^T + bo
    gemm_xwT<<<gG, blk, 0, stream>>>(ctx, Wo,     bo,      nullptr, out, EE, EE);
}